// SAGENet_41291815584512
// MI455X (gfx1250) — compile-verified
//
#include <hip/hip_runtime.h>

typedef __attribute__((ext_vector_type(2))) float v2f;
typedef __attribute__((ext_vector_type(8))) float v8f;

// ---------------------------------------------------------------------------
// Zero-init workspace region (graph-capture safe, run every call)
// ---------------------------------------------------------------------------
__global__ void zero_kernel(float* __restrict__ p, long long n) {
  long long t = (long long)blockIdx.x * blockDim.x + threadIdx.x;
  if (t < n) p[t] = 0.0f;
}

// ---------------------------------------------------------------------------
// Edge scatter: agg[dst][b][c] += ew[e] * X[src][b][c];  deg[dst] += 1
// 16 threads per edge -> one (b, c-quad) each: float4 gather + 4 f32 atomics.
// Edge arrays are read-once -> non-temporal loads (keep the reused gather
// source + accumulator resident in the 192MB L2). Atomics are L2 RMW.
// ---------------------------------------------------------------------------
__global__ void sage_scatter_kernel(const float* __restrict__ X,
                                    const float* __restrict__ ew,
                                    const int* __restrict__ src,
                                    const int* __restrict__ dst,
                                    float* __restrict__ agg,
                                    float* __restrict__ deg,
                                    long long nE,
                                    int x_node_stride,
                                    int x_b_stride) {
  long long t = (long long)blockIdx.x * blockDim.x + threadIdx.x;
  long long e = t >> 4;
  if (e >= nE) return;
  int i  = (int)(t & 15);
  int b  = i >> 2;         // batch index
  int c4 = (i & 3) << 2;   // column quad base

  int   s = __builtin_nontemporal_load(&src[e]);
  int   d = __builtin_nontemporal_load(&dst[e]);
  float w = __builtin_nontemporal_load(&ew[e]);

  const float4 xv =
      *(const float4*)(X + (long long)s * x_node_stride + (long long)b * x_b_stride + c4);

  float* ap = agg + (((long long)d << 6) + (b << 4) + c4);
  unsafeAtomicAdd(ap + 0, w * xv.x);
  unsafeAtomicAdd(ap + 1, w * xv.y);
  unsafeAtomicAdd(ap + 2, w * xv.z);
  unsafeAtomicAdd(ap + 3, w * xv.w);
  if (i == 0) unsafeAtomicAdd(&deg[d], 1.0f);
}

// ---------------------------------------------------------------------------
// Fused finalize: h = [X[res_n_id] | agg/max(deg,1)]  (16x32 per 4 nodes)
//                 out = leaky_relu(h @ W + bias)
// One wave handles 4 dst nodes: M = 16 rows = 4 nodes x 4 batch.
// GEMM via 8 chained V_WMMA_F32_16X16X4_F32 (full fp32, K=32 total).
//
// A-frag (16x4 f32, 2 VGPRs): lane L -> row L%16, K columns 2*(L/16), +1
// B-frag (4x16 f32, 2 VGPRs): lane L -> col L%16, K rows   2*(L/16), +1
// D (16x16 f32, 8 VGPRs):     lane L, vgpr r -> [r + 8*(L/16)][L%16]
// All strides fit in 32 bits -> int indexing (base + voffset addressing).
// ---------------------------------------------------------------------------
__global__ void sage_finalize_wmma_kernel(const float* __restrict__ Xin,
                                          const float* __restrict__ agg,
                                          const float* __restrict__ deg,
                                          const int* __restrict__ resid,
                                          const float* __restrict__ W,     // [32,16] row-major
                                          const float* __restrict__ bias,  // [16]
                                          float* __restrict__ out,
                                          int nDst,
                                          int x_node_stride,
                                          int x_b_stride,
                                          int o_node_stride,
                                          int o_b_stride,
                                          float neg_slope) {
  int lane = threadIdx.x & 31;
  int wave = threadIdx.x >> 5;
  int nodeBase = (blockIdx.x * (blockDim.x >> 5) + wave) * 4;
  if (nodeBase >= nDst) return;  // wave-uniform: EXEC stays all-ones for WMMA

  int rowA  = lane & 15;   // M row this lane feeds for the A fragment
  int khalf = lane >> 4;   // which K half-pair this lane holds
  int q = rowA >> 2;       // node within the quad
  int b = rowA & 3;        // batch index
  int node = nodeBase + q;

  int   res    = resid[node];
  float dv     = deg[node];
  float invdeg = 1.0f / fmaxf(dv, 1.0f);

  const float* xRow   = Xin + res * x_node_stride + b * x_b_stride;
  const float* aggRow = agg + (node << 6) + (b << 4);

  int col = lane & 15;     // N column this lane owns for B and D fragments
  v8f acc = {};

#pragma unroll
  for (int k = 0; k < 8; ++k) {
    int c0 = 4 * k + 2 * khalf;  // K column pair for this lane
    v2f a, bf;
    if (k < 4) {
      // first 16 concat columns: X[res_n_id[node]]
      a.x = xRow[c0];
      a.y = xRow[c0 + 1];
    } else {
      // last 16 concat columns: mean aggregation
      a.x = aggRow[c0 - 16] * invdeg;
      a.y = aggRow[c0 - 15] * invdeg;
    }
    bf.x = W[(c0) * 16 + col];
    bf.y = W[(c0 + 1) * 16 + col];
    acc = __builtin_amdgcn_wmma_f32_16x16x4_f32(
        /*neg_a=*/false, a, /*neg_b=*/false, bf,
        /*c_mod=*/(short)0, acc, /*reuse_a=*/false, /*reuse_b=*/false);
  }

  float bv = bias[col];
#pragma unroll
  for (int r = 0; r < 8; ++r) {
    int M = r + (khalf << 3);
    float v = acc[r] + bv;
    v = (v > 0.0f) ? v : v * neg_slope;  // leaky_relu
    int nOut = nodeBase + (M >> 2);
    int bb   = M & 3;
    out[nOut * o_node_stride + bb * o_b_stride + col] = v;
  }
}

// ---------------------------------------------------------------------------
// Launch
// ---------------------------------------------------------------------------
extern "C" void kernel_launch(void* const* d_in, const int* in_sizes, int n_in,
                              void* d_out, int out_size, void* d_ws, size_t ws_size,
                              hipStream_t stream) {
  const float* X   = (const float*)d_in[0];   // [B=4, N1, C=16]
  const float* W1  = (const float*)d_in[1];   // [32,16]
  const float* b1  = (const float*)d_in[2];   // [16]
  const float* W2  = (const float*)d_in[3];   // [32,16]
  const float* b2  = (const float*)d_in[4];   // [16]
  const float* ew1 = (const float*)d_in[5];   // [E1]
  const float* ew2 = (const float*)d_in[6];   // [E2]
  const int* src1  = (const int*)d_in[7];
  const int* dst1  = (const int*)d_in[8];
  const int* src2  = (const int*)d_in[9];
  const int* dst2  = (const int*)d_in[10];
  const int* res1  = (const int*)d_in[11];
  const int* res2  = (const int*)d_in[12];

  const int BATCH = 4, C = 16;
  const long long E1 = in_sizes[5];
  const long long E2 = in_sizes[6];
  const int ND1      = in_sizes[11];
  const int ND2      = in_sizes[12];
  const int N1       = in_sizes[0] / (BATCH * C);

  // workspace layout: agg1 | deg1 | agg2 | deg2 | h1   (floats)
  float* ws   = (float*)d_ws;
  float* agg1 = ws;
  float* deg1 = agg1 + (long long)ND1 * 64;
  float* agg2 = deg1 + ND1;
  float* deg2 = agg2 + (long long)ND2 * 64;
  float* h1   = deg2 + ND2;

  const int THR = 256;

  // zero accumulators (agg1|deg1|agg2|deg2 are contiguous)
  long long zeroN = (long long)ND1 * 64 + ND1 + (long long)ND2 * 64 + ND2;
  zero_kernel<<<(unsigned)((zeroN + THR - 1) / THR), THR, 0, stream>>>(ws, zeroN);

  // ---- layer 1 ----
  long long t1 = E1 * 16;
  sage_scatter_kernel<<<(unsigned)((t1 + THR - 1) / THR), THR, 0, stream>>>(
      X, ew1, src1, dst1, agg1, deg1, E1,
      /*x_node_stride=*/C, /*x_b_stride=*/N1 * C);

  int blocks1 = (ND1 / 4 + 7) / 8;  // 8 waves per block, 4 nodes per wave
  sage_finalize_wmma_kernel<<<blocks1, 256, 0, stream>>>(
      X, agg1, deg1, res1, W1, b1, h1, ND1,
      /*x_node_stride=*/C, /*x_b_stride=*/N1 * C,
      /*o_node_stride=*/64, /*o_b_stride=*/16, 0.01f);

  // ---- layer 2 (input h1 is [node][b][c]) ----
  long long t2 = E2 * 16;
  sage_scatter_kernel<<<(unsigned)((t2 + THR - 1) / THR), THR, 0, stream>>>(
      h1, ew2, src2, dst2, agg2, deg2, E2,
      /*x_node_stride=*/64, /*x_b_stride=*/16);

  int blocks2 = (ND2 / 4 + 7) / 8;
  // final output written directly transposed: out[b][n][c]
  sage_finalize_wmma_kernel<<<blocks2, 256, 0, stream>>>(
      h1, agg2, deg2, res2, W2, b2, (float*)d_out, ND2,
      /*x_node_stride=*/64, /*x_b_stride=*/16,
      /*o_node_stride=*/16, /*o_b_stride=*/ND2 * 16, 0.01f);
}